// AttrGeneration_47510928228698
// MI455X (gfx1250) — compile-verified
//
#include <hip/hip_runtime.h>

#define N_NODES 200000
#define F_DIM   512
#define L_DIM   40
#define H_DIM   256
#define SG_DIM  512
#define LEAK    0.01f
#define GDIM    (2 * F_DIM + 3 * L_DIM)   // 1144

typedef __attribute__((ext_vector_type(2))) float v2f;
typedef __attribute__((ext_vector_type(4))) float floatx4;
typedef __attribute__((ext_vector_type(8))) float v8f;

// ---------------------------------------------------------------------------
// 1) Streaming copy of feat -> out rows [0, N). 820 MB of traffic: this IS the
//    kernel's runtime (~35us at 23.3 TB/s). float4 + non-temporal hints (no
//    reuse; 410MB working set exceeds the 192MB L2).
// ---------------------------------------------------------------------------
__global__ void copy_feat_kernel(const floatx4* __restrict__ src,
                                 floatx4* __restrict__ dst, long long n4) {
  long long i = (long long)blockIdx.x * blockDim.x + threadIdx.x;
  if (i < n4) {
    floatx4 v = __builtin_nontemporal_load(&src[i]);
    __builtin_nontemporal_store(v, &dst[i]);
  }
}

// ---------------------------------------------------------------------------
// 2) sub_graph_emb = mean over 512 gathered rows of node_emb. Coalesced along
//    the feature dimension (thread = column).
// ---------------------------------------------------------------------------
__global__ void subgraph_mean_kernel(const int* __restrict__ nodes,
                                     const float* __restrict__ emb,
                                     float* __restrict__ out) {
  int col = blockIdx.x * blockDim.x + threadIdx.x;   // 0..511
  float s = 0.0f;
  for (int i = 0; i < SG_DIM; ++i) {
    s += emb[(long long)nodes[i] * F_DIM + col];
  }
  out[col] = s * (1.0f / (float)SG_DIM);
}

// ---------------------------------------------------------------------------
// 3) Generic GEMV via V_WMMA_F32_16X16X4_F32 (full fp32 precision).
//    One wave (32 threads) per 16 outputs. A-matrix 16x4 layout: row M=0 of A
//    carries the input vector -> lane 0 holds (K=0,K=1), lane 16 holds
//    (K=2,K=3); all other lanes zero. B-matrix 4x16: VGPR0 = rows K=0/K=2
//    striped across lane halves, VGPR1 = rows K=1/K=3. Result row M=0 lands
//    in acc[0], lanes 0..15. act: 0=none, 1=relu, 2=leaky(0.01).
// ---------------------------------------------------------------------------
__global__ void gemv_wmma_kernel(const float* __restrict__ x,
                                 const int* __restrict__ row_idx,
                                 const float* __restrict__ W,
                                 const float* __restrict__ bias,
                                 float* __restrict__ y,
                                 int K, int Nout, int act) {
  const int lane = threadIdx.x;        // 0..31 (one wave; EXEC all ones)
  const int half = lane >> 4;          // 0 or 1
  const int col  = lane & 15;
  const int n0   = blockIdx.x * 16;

  if (row_idx) x += (long long)row_idx[0] * K;

  int n = n0 + col;
  if (n > Nout - 1) n = Nout - 1;      // clamp (edge tile loads stay in-bounds)

  v8f acc = {0.f, 0.f, 0.f, 0.f, 0.f, 0.f, 0.f, 0.f};
  for (int k = 0; k < K; k += 4) {     // K is a multiple of 4 for all calls
    float a0 = x[k + 2 * half];
    float a1 = x[k + 2 * half + 1];
    a0 = (col == 0) ? a0 : 0.0f;       // only row M=0 of A is live
    a1 = (col == 0) ? a1 : 0.0f;
    v2f A; A[0] = a0; A[1] = a1;
    v2f B;
    B[0] = W[(long long)(k + 2 * half) * Nout + n];
    B[1] = W[(long long)(k + 2 * half + 1) * Nout + n];
    acc = __builtin_amdgcn_wmma_f32_16x16x4_f32(
        /*neg_a=*/false, A, /*neg_b=*/false, B,
        /*c_mod=*/(short)0, acc, /*reuse_a=*/false, /*reuse_b=*/false);
  }

  if (lane < 16 && (n0 + lane) < Nout) {
    float r = acc[0];                  // (M=0, N=n0+lane)
    if (bias) r += bias[n0 + lane];
    if (act == 1)      r = fmaxf(r, 0.0f);
    else if (act == 2) r = (r >= 0.0f) ? r : (LEAK * r);
    y[n0 + lane] = r;
  }
}

// ---------------------------------------------------------------------------
// 4) graph_emb = concat([sub_emb, node_emb[target], tarfeat, wlabel, wsec])
// ---------------------------------------------------------------------------
__global__ void assemble_kernel(const float* __restrict__ sub_emb,
                                const float* __restrict__ node_emb,
                                const int* __restrict__ target,
                                const float* __restrict__ tarfeat,
                                const float* __restrict__ wlabel,
                                const float* __restrict__ wsec,
                                float* __restrict__ graph) {
  long long tbase = (long long)target[0] * F_DIM;
  for (int i = threadIdx.x; i < GDIM; i += blockDim.x) {
    float v;
    if (i < F_DIM)                    v = sub_emb[i];
    else if (i < 2 * F_DIM)           v = node_emb[tbase + (i - F_DIM)];
    else if (i < 2 * F_DIM + L_DIM)   v = tarfeat[i - 2 * F_DIM];
    else if (i < 2 * F_DIM + 2*L_DIM) v = wlabel[i - 2 * F_DIM - L_DIM];
    else                              v = wsec[i - 2 * F_DIM - 2 * L_DIM];
    graph[i] = v;
  }
}

// ---------------------------------------------------------------------------
// 5) gumbel_topk (eval path): tmp0 = softmax(x*1000); repeat budget-1 times:
//    argmax(current), mask[idx]=9999, re-softmax, accumulate. LDS tree
//    reductions; argmax tie-break = lowest index (matches jnp.argmax).
//    Writes inj_feat both as new_feat row N and as the second tuple output.
// ---------------------------------------------------------------------------
__global__ void topk_kernel(const float* __restrict__ add_feat,
                            const int* __restrict__ budget_p,
                            float* __restrict__ out) {
  __shared__ float sv[F_DIM];
  __shared__ int   si[F_DIM];
  __shared__ int   bidx;
  const int tid    = threadIdx.x;      // 512 threads
  const int budget = budget_p[0];
  const float logit = add_feat[tid];
  float my_mask = 0.0f;
  float acc     = 0.0f;

  for (int it = 0; it < budget; ++it) {
    if (it > 0) {
      // argmax of current masked logits (== argmax of current softmax)
      float l = logit - my_mask;
      sv[tid] = l; si[tid] = tid; __syncthreads();
      for (int off = F_DIM / 2; off > 0; off >>= 1) {
        if (tid < off) {
          float vo = sv[tid + off]; int io = si[tid + off];
          if (vo > sv[tid] || (vo == sv[tid] && io < si[tid])) {
            sv[tid] = vo; si[tid] = io;
          }
        }
        __syncthreads();
      }
      if (tid == 0) bidx = si[0];
      __syncthreads();
      if (tid == bidx) my_mask = 9999.0f;
    }

    float l = logit - my_mask;
    // block max
    sv[tid] = l; __syncthreads();
    for (int off = F_DIM / 2; off > 0; off >>= 1) {
      if (tid < off) sv[tid] = fmaxf(sv[tid], sv[tid + off]);
      __syncthreads();
    }
    float m = sv[0]; __syncthreads();
    float e = expf((l - m) * 1000.0f);   // scale = 0.01*tau = 0.001
    // block sum
    sv[tid] = e; __syncthreads();
    for (int off = F_DIM / 2; off > 0; off >>= 1) {
      if (tid < off) sv[tid] += sv[tid + off];
      __syncthreads();
    }
    float s = sv[0]; __syncthreads();
    acc += e / s;
  }

  out[(long long)N_NODES * F_DIM + tid]       = acc;  // new_feat row N
  out[(long long)(N_NODES + 1) * F_DIM + tid] = acc;  // tuple output #2
}

// ---------------------------------------------------------------------------
extern "C" void kernel_launch(void* const* d_in, const int* in_sizes, int n_in,
                              void* d_out, int out_size, void* d_ws, size_t ws_size,
                              hipStream_t stream) {
  const int*   target  = (const int*)d_in[0];
  const int*   sgn     = (const int*)d_in[1];
  const float* feat    = (const float*)d_in[2];
  const float* nemb    = (const float*)d_in[3];
  const float* wlabel  = (const float*)d_in[4];
  const float* wsec    = (const float*)d_in[5];
  const float* weight1 = (const float*)d_in[6];
  const float* weight2 = (const float*)d_in[7];
  const float* W1      = (const float*)d_in[8];
  const float* b1      = (const float*)d_in[9];
  const float* W2      = (const float*)d_in[10];
  const float* b2      = (const float*)d_in[11];
  const float* W3      = (const float*)d_in[12];
  const float* b3      = (const float*)d_in[13];
  const int*   fnum    = (const int*)d_in[14];

  float* out = (float*)d_out;
  float* ws  = (float*)d_ws;
  float* sub_emb = ws;          // 512
  float* tmp_emb = ws + 1024;   // 256
  float* tarfeat = ws + 2048;   // 40
  float* graph   = ws + 3072;   // 1144
  float* h1      = ws + 5120;   // 128
  float* h2      = ws + 6144;   // 512
  float* addf    = ws + 7168;   // 512

  // (A) The bandwidth-bound bulk copy (independent of the tiny compute chain).
  long long n4 = (long long)N_NODES * F_DIM / 4;        // 25.6M float4
  int threads = 256;
  unsigned blocks = (unsigned)((n4 + threads - 1) / threads);
  copy_feat_kernel<<<blocks, threads, 0, stream>>>(
      (const floatx4*)feat, (floatx4*)out, n4);

  // (B) Tiny compute chain, sequential on the stream.
  subgraph_mean_kernel<<<2, 256, 0, stream>>>(sgn, nemb, sub_emb);
  // tmp_emb = relu(feat[target] @ weight1)           [512 -> 256]
  gemv_wmma_kernel<<<H_DIM / 16, 32, 0, stream>>>(
      feat, target, weight1, nullptr, tmp_emb, F_DIM, H_DIM, 1);
  // tarfeat = tmp_emb @ weight2                      [256 -> 40]
  gemv_wmma_kernel<<<(L_DIM + 15) / 16, 32, 0, stream>>>(
      tmp_emb, nullptr, weight2, nullptr, tarfeat, H_DIM, L_DIM, 0);
  assemble_kernel<<<1, 256, 0, stream>>>(
      sub_emb, nemb, target, tarfeat, wlabel, wsec, graph);
  // h1 = leaky(graph @ W1 + b1)                      [1144 -> 128]
  gemv_wmma_kernel<<<128 / 16, 32, 0, stream>>>(
      graph, nullptr, W1, b1, h1, GDIM, 128, 2);
  // h2 = leaky(h1 @ W2 + b2)                         [128 -> 512]
  gemv_wmma_kernel<<<512 / 16, 32, 0, stream>>>(
      h1, nullptr, W2, b2, h2, 128, 512, 2);
  // add_feat = h2 @ W3 + b3                          [512 -> 512]
  gemv_wmma_kernel<<<512 / 16, 32, 0, stream>>>(
      h2, nullptr, W3, b3, addf, 512, 512, 0);
  // (C) top-k accumulate + write inj_feat (row N and tuple output #2)
  topk_kernel<<<1, F_DIM, 0, stream>>>(addf, fnum, out);
}